// GNN_LP_54357106098297
// MI455X (gfx1250) — compile-verified
//
#include <hip/hip_runtime.h>
#include <math.h>

#define HID 128
#define NNODES 100000

typedef float v2f __attribute__((ext_vector_type(2)));
typedef float v8f __attribute__((ext_vector_type(8)));

__global__ void zero_f32(float* __restrict__ p, int n) {
  int i = blockIdx.x * blockDim.x + threadIdx.x;
  if (i < n) p[i] = 0.0f;
}

__global__ void count_deg(const int* __restrict__ dst, float* __restrict__ deg, int nE) {
  int e = blockIdx.x * blockDim.x + threadIdx.x;
  if (e < nE) atomicAdd(&deg[dst[e]], 1.0f);
}

__global__ void finalize_inv_deg(float* __restrict__ deg, int n) {
  int i = blockIdx.x * blockDim.x + threadIdx.x;
  if (i < n) deg[i] = 1.0f / fmaxf(deg[i], 1.0f);
}

// One wave (32 lanes) per edge: coalesced 512B gather of h[src] row,
// 4 atomicAdds per lane into agg[dst] (agg stays resident in 192MB L2).
__global__ void scatter_rows(const float* __restrict__ h,
                             const int* __restrict__ src,
                             const int* __restrict__ dst,
                             float* __restrict__ agg, int nE) {
  int wid = (blockIdx.x * blockDim.x + threadIdx.x) >> 5;
  int lane = threadIdx.x & 31;
  if (wid >= nE) return;
  int s = src[wid];
  int d = dst[wid];
  const float4* hrow = (const float4*)(h + (size_t)s * HID);
  float4 v = hrow[lane];
  float* arow = agg + (size_t)d * HID + (size_t)lane * 4;
  atomicAdd(arow + 0, v.x);
  atomicAdd(arow + 1, v.y);
  atomicAdd(arow + 2, v.z);
  atomicAdd(arow + 3, v.w);
}

// h_out = x @ W  for x[N,128], W[128,128].
// Block = 256 threads = 8 waves; block owns a 16-row strip, wave w owns
// the 16-col tile starting at 16*w. FP32 WMMA 16x16x4, 32 K-steps split
// across two independent accumulator chains for ILP.
__global__ __launch_bounds__(256) void enc_gemm(const float* __restrict__ A,
                                                const float* __restrict__ W,
                                                float* __restrict__ out) {
  const int rowBase = blockIdx.x * 16;
  const int wave = threadIdx.x >> 5;
  const int lane = threadIdx.x & 31;
  const int colBase = wave * 16;
  const int m = lane & 15;
  const int half = lane >> 4;   // 0: K=0,1  1: K=2,3 within each 4-wide step
  const size_t row = (size_t)rowBase + m;
  const float* arow = A + row * HID;
  v8f acc0 = {}, acc1 = {};
#pragma unroll
  for (int k = 0; k < HID; k += 8) {
    {
      const int ka = k + half * 2;
      v2f a = *(const v2f*)(arow + ka);           // A[m][ka], A[m][ka+1]
      v2f b;
      b.x = W[(size_t)ka * HID + colBase + m];    // B[ka][n]
      b.y = W[(size_t)(ka + 1) * HID + colBase + m];
      acc0 = __builtin_amdgcn_wmma_f32_16x16x4_f32(false, a, false, b,
                                                   (short)0, acc0, false, false);
    }
    {
      const int ka = k + 4 + half * 2;
      v2f a = *(const v2f*)(arow + ka);
      v2f b;
      b.x = W[(size_t)ka * HID + colBase + m];
      b.y = W[(size_t)(ka + 1) * HID + colBase + m];
      acc1 = __builtin_amdgcn_wmma_f32_16x16x4_f32(false, a, false, b,
                                                   (short)0, acc1, false, false);
    }
  }
  v8f acc = acc0 + acc1;
#pragma unroll
  for (int v = 0; v < 8; ++v) {
    size_t r = (size_t)rowBase + half * 8 + v;  // C[v + half*8][m]
    out[r * HID + colBase + m] = acc[v];
  }
}

// out = h + relu( l2norm( (agg*inv_deg)@Wl + bl + h@Wr ) )
// Block = 16-row strip; the two GEMMs run on independent accumulator chains;
// per-row L2 norm reduced wave-locally with shfl_xor, then across the 8
// waves through LDS.
__global__ __launch_bounds__(256) void sage_combine(
    const float* __restrict__ h, const float* __restrict__ agg,
    const float* __restrict__ invdeg,
    const float* __restrict__ Wl, const float* __restrict__ Wr,
    const float* __restrict__ bl, float* __restrict__ out) {
  __shared__ float s_ssq[8][16];
  __shared__ float s_inv[16];
  const int rowBase = blockIdx.x * 16;
  const int wave = threadIdx.x >> 5;
  const int lane = threadIdx.x & 31;
  const int colBase = wave * 16;
  const int m = lane & 15;
  const int half = lane >> 4;
  const size_t row = (size_t)rowBase + m;
  const float idg = invdeg[row];
  const float* hrow = h + row * HID;
  const float* grow = agg + row * HID;
  v8f accL = {}, accR = {};
#pragma unroll 8
  for (int k = 0; k < HID; k += 4) {
    const int ka = k + half * 2;
    {
      v2f a = *(const v2f*)(grow + ka);
      a.x *= idg;
      a.y *= idg;
      v2f b;
      b.x = Wl[(size_t)ka * HID + colBase + m];
      b.y = Wl[(size_t)(ka + 1) * HID + colBase + m];
      accL = __builtin_amdgcn_wmma_f32_16x16x4_f32(false, a, false, b,
                                                   (short)0, accL, false, false);
    }
    {
      v2f a = *(const v2f*)(hrow + ka);
      v2f b;
      b.x = Wr[(size_t)ka * HID + colBase + m];
      b.y = Wr[(size_t)(ka + 1) * HID + colBase + m];
      accR = __builtin_amdgcn_wmma_f32_16x16x4_f32(false, a, false, b,
                                                   (short)0, accR, false, false);
    }
  }
  const float bias = bl[colBase + m];
  v8f acc = accL + accR;
  float rs[8];
#pragma unroll
  for (int v = 0; v < 8; ++v) {
    acc[v] += bias;
    float sq = acc[v] * acc[v];
    sq += __shfl_xor(sq, 1, 32);   // reduce over the 16 lanes holding one row
    sq += __shfl_xor(sq, 2, 32);
    sq += __shfl_xor(sq, 4, 32);
    sq += __shfl_xor(sq, 8, 32);
    rs[v] = sq;                    // row sum for local row (half*8 + v)
  }
  if (m == 0) {
#pragma unroll
    for (int v = 0; v < 8; ++v) s_ssq[wave][half * 8 + v] = rs[v];
  }
  __syncthreads();
  if (threadIdx.x < 16) {
    float t = 0.0f;
#pragma unroll
    for (int w = 0; w < 8; ++w) t += s_ssq[w][threadIdx.x];
    s_inv[threadIdx.x] = 1.0f / fmaxf(sqrtf(t), 1e-12f);
  }
  __syncthreads();
#pragma unroll
  for (int v = 0; v < 8; ++v) {
    const int r = half * 8 + v;
    float o = fmaxf(acc[v] * s_inv[r], 0.0f);            // relu(normalized)
    size_t gr = (size_t)rowBase + r;
    out[gr * HID + colBase + m] = h[gr * HID + colBase + m] + o;  // residual
  }
}

extern "C" void kernel_launch(void* const* d_in, const int* in_sizes, int n_in,
                              void* d_out, int out_size, void* d_ws, size_t ws_size,
                              hipStream_t stream) {
  const float* x    = (const float*)d_in[0];
  const int*   ei   = (const int*)d_in[1];
  const float* encW = (const float*)d_in[2];
  const float* Wl   = (const float*)d_in[3];
  const float* bl   = (const float*)d_in[4];
  const float* Wr   = (const float*)d_in[5];

  const int E = in_sizes[1] / 2;        // edge_index is [2, E]
  const int* src = ei;
  const int* dst = ei + E;
  const int N = NNODES;
  const int NH = N * HID;

  // Workspace: one feature ping-pong buffer + agg + inv_deg (~103 MB).
  float* hbuf = (float*)d_ws;
  float* agg  = hbuf + (size_t)NH;
  float* invd = agg + (size_t)NH;
  float* outb = (float*)d_out;

  // Degrees -> inverse degrees.
  zero_f32<<<(N + 255) / 256, 256, 0, stream>>>(invd, N);
  count_deg<<<(E + 255) / 256, 256, 0, stream>>>(dst, invd, E);
  finalize_inv_deg<<<(N + 255) / 256, 256, 0, stream>>>(invd, N);

  // Encoder: h = x @ enc_W  (N divisible by 16).
  enc_gemm<<<N / 16, 256, 0, stream>>>(x, encW, hbuf);

  const float* cur = hbuf;
  for (int l = 0; l < 3; ++l) {
    zero_f32<<<(NH + 255) / 256, 256, 0, stream>>>(agg, NH);
    scatter_rows<<<(int)(((size_t)E * 32 + 255) / 256), 256, 0, stream>>>(
        cur, src, dst, agg, E);
    float* outp = (l & 1) ? hbuf : outb;   // layer 0 -> d_out, 1 -> ws, 2 -> d_out
    sage_combine<<<N / 16, 256, 0, stream>>>(
        cur, agg, invd,
        Wl + (size_t)l * HID * HID, Wr + (size_t)l * HID * HID,
        bl + (size_t)l * HID, outp);
    cur = outp;
  }
}